// GraphNeuralNetwork_75831942578635
// MI455X (gfx1250) — compile-verified
//
#include <hip/hip_runtime.h>
#include <math.h>

// Problem constants (match reference)
#define B_     128
#define NE_    64
#define NN_    16
#define D_     128
#define NRBF_  16
#define L_     3
#define NUP_   32
#define CUTOFF_ 2.0f
#define EPS_    1e-12f

#define KEE_   (NE_ * NRBF_)   // 1024
#define KNE_   (NN_ * NRBF_)   // 256
#define NT_    4               // N-tiles per wave (16*NT = 64-wide strip)
#define WSZ_   (D_ * D_)       // 16384 floats per weight matrix

typedef float v2f __attribute__((ext_vector_type(2)));
typedef float v8f __attribute__((ext_vector_type(8)));

// ---------------------------------------------------------------------------
// f32 WMMA 16x16x4 helpers (wave32; one wave owns a 16 x (16*NT) strip).
// A fragment (16x4, 32-bit): lanes 0-15 -> M=lane,    VGPR0/1 = K=k0,k0+1
//                            lanes 16-31 -> M=lane-16, VGPR0/1 = K=k0+2,k0+3
// B fragment (4x16): lanes 0-15 -> N=lane (rows k0,k0+1); lanes 16-31 -> rows k0+2,k0+3
// C/D: VGPR i: lanes 0-15 -> (M=i, N=lane); lanes 16-31 -> (M=8+i, N=lane-16)
//
// Dense-GEMM B operands are pre-packed row-pair interleaved:
//   Wp[(k>>1)*2*D + col*2 + (k&1)] = W[k*D + col]
// so the (ka, ka+1) fragment is one aligned 8-byte load (ka is always even).
// ---------------------------------------------------------------------------
__device__ __forceinline__ void gemm_strip_f32p(const float* __restrict__ A, int lda,
                                                const float* __restrict__ Bp,  // packed
                                                int m0, int n0, v8f acc[NT_], int lane) {
  const int half = lane >> 4;
  const int l16  = lane & 15;
  const float* __restrict__ arow = A + (size_t)(m0 + l16) * lda;
#pragma unroll
  for (int k0 = 0; k0 < D_; k0 += 4) {
    const int ka = k0 + half * 2;               // even
    const v2f a = *(const v2f*)&arow[ka];
    const float* __restrict__ brow = Bp + (size_t)(ka >> 1) * (2 * D_);
#pragma unroll
    for (int t = 0; t < NT_; ++t) {
      const v2f b = *(const v2f*)&brow[(n0 + t * 16 + l16) * 2];
      acc[t] = __builtin_amdgcn_wmma_f32_16x16x4_f32(false, a, false, b, (short)0,
                                                     acc[t], false, false);
    }
  }
}

__device__ __forceinline__ void store_strip_f32(float* __restrict__ C, int ldc,
                                                int m0, int n0, v8f acc[NT_], int lane) {
  const int half = lane >> 4;
  const int l16  = lane & 15;
#pragma unroll
  for (int t = 0; t < NT_; ++t)
#pragma unroll
    for (int ii = 0; ii < 8; ++ii)
      C[(size_t)(m0 + half * 8 + ii) * ldc + n0 + t * 16 + l16] = acc[t][ii];
}

__device__ __forceinline__ void load_strip_f32(const float* __restrict__ C, int ldc,
                                               int m0, int n0, v8f acc[NT_], int lane) {
  const int half = lane >> 4;
  const int l16  = lane & 15;
#pragma unroll
  for (int t = 0; t < NT_; ++t)
#pragma unroll
    for (int ii = 0; ii < 8; ++ii)
      acc[t][ii] = C[(size_t)(m0 + half * 8 + ii) * ldc + n0 + t * 16 + l16];
}

// ---------------------------------------------------------------------------
// K-pack: interleave row pairs of the 12 dense weight matrices (4 per layer):
//   slot 0: W_msg_ee[l], 1: W_msg_ne[l], 2: W1[l], 3: W2[l]
// ---------------------------------------------------------------------------
__global__ void gnn_pack_w(const float* __restrict__ We, const float* __restrict__ Wn,
                           const float* __restrict__ W1, const float* __restrict__ W2,
                           float* __restrict__ Wp) {
  const int gid = blockIdx.x * blockDim.x + threadIdx.x;   // 12 * 16384 total
  const int m   = gid >> 14;           // matrix slot 0..11
  const int idx = gid & (WSZ_ - 1);
  const int l = m >> 2, w = m & 3;
  const float* src = (w == 0 ? We : w == 1 ? Wn : w == 2 ? W1 : W2) + (size_t)l * WSZ_;
  const int k = idx >> 7, col = idx & (D_ - 1);
  Wp[(size_t)m * WSZ_ + (size_t)(k >> 1) * (2 * D_) + col * 2 + (k & 1)] = src[idx];
}

// ---------------------------------------------------------------------------
// K0: feat_ee -> GEMM-A layout [b][i][(j*NRBF+k)]  (8.4M floats)
// ---------------------------------------------------------------------------
__global__ void gnn_feat_ee(const float* __restrict__ r, float* __restrict__ Aee) {
  const int bi = blockIdx.x;            // b*NE + i
  const int b = bi >> 6, i = bi & 63;
  const float rx = r[(size_t)(b * NE_ + i) * 3 + 0];
  const float ry = r[(size_t)(b * NE_ + i) * 3 + 1];
  const float rz = r[(size_t)(b * NE_ + i) * 3 + 2];
  const float inv_cut = 1.0f / CUTOFF_;
  const float dmu     = CUTOFF_ / (float)(NRBF_ - 1);
  const float inv_sig = (float)NRBF_ / CUTOFF_;
  for (int t = threadIdx.x; t < KEE_; t += blockDim.x) {
    const int j = t >> 4, k = t & 15;
    const float dx = rx - r[(size_t)(b * NE_ + j) * 3 + 0];
    const float dy = ry - r[(size_t)(b * NE_ + j) * 3 + 1];
    const float dz = rz - r[(size_t)(b * NE_ + j) * 3 + 2];
    const float dist = sqrtf(dx * dx + dy * dy + dz * dz + EPS_);
    float env = 1.0f - dist * inv_cut;
    env = (dist < CUTOFF_) ? env * env : 0.0f;
    const float z = (dist - dmu * (float)k) * inv_sig;
    float g = expf(-0.5f * z * z) * env;
    if (i == j) g = 0.0f;
    Aee[(size_t)bi * KEE_ + t] = g;
  }
}

// ---------------------------------------------------------------------------
// K1: feat_ne -> GEMM-A layout [b][j][(a*NRBF+k)]  (2.1M floats)
// ---------------------------------------------------------------------------
__global__ void gnn_feat_ne(const float* __restrict__ r, const float* __restrict__ coords,
                            float* __restrict__ Ane) {
  const int bj = blockIdx.x;            // b*NE + j
  const int b = bj >> 6, j = bj & 63;
  const float rx = r[(size_t)(b * NE_ + j) * 3 + 0];
  const float ry = r[(size_t)(b * NE_ + j) * 3 + 1];
  const float rz = r[(size_t)(b * NE_ + j) * 3 + 2];
  const float inv_cut = 1.0f / CUTOFF_;
  const float dmu     = CUTOFF_ / (float)(NRBF_ - 1);
  const float inv_sig = (float)NRBF_ / CUTOFF_;
  const int t = threadIdx.x;            // 256 threads = NN*NRBF
  const int a = t >> 4, k = t & 15;
  const float dx = rx - coords[a * 3 + 0];
  const float dy = ry - coords[a * 3 + 1];
  const float dz = rz - coords[a * 3 + 2];
  const float dist = sqrtf(dx * dx + dy * dy + dz * dz + EPS_);
  float env = 1.0f - dist * inv_cut;
  env = (dist < CUTOFF_) ? env * env : 0.0f;
  const float z = (dist - dmu * (float)k) * inv_sig;
  const float g = expf(-0.5f * z * z) * env;
  Ane[(size_t)bj * KNE_ + t] = g;
}

// ---------------------------------------------------------------------------
// K2: h init from spin embeddings
// ---------------------------------------------------------------------------
__global__ void gnn_init_h(const float* __restrict__ spin_emb, float* __restrict__ h) {
  const int gid = blockIdx.x * blockDim.x + threadIdx.x;   // B*NE*D total
  const int d = gid & (D_ - 1);
  const int i = (gid >> 7) & (NE_ - 1);
  h[gid] = spin_emb[(i < NUP_ ? 0 : 1) * D_ + d];
}

// ---------------------------------------------------------------------------
// K3: hW = h @ W_msg_ee[l]  (8192x128 @ 128x128) and hnW = nuc_emb @ W_msg_ne[l]
// One wave per 16x64 strip. Strips 0..1023: hW, 1024..1025: hnW.
// ---------------------------------------------------------------------------
__global__ void gnn_hw(const float* __restrict__ h, const float* __restrict__ nuc_emb,
                       const float* __restrict__ Wp_ee, const float* __restrict__ Wp_ne,
                       float* __restrict__ hW, float* __restrict__ hnW) {
  const int wid = blockIdx.x;
  const int lane = threadIdx.x & 31;
  v8f acc[NT_] = {};
  if (wid < 1024) {
    const int m0 = (wid >> 1) * 16;     // over B*NE = 8192 rows
    const int n0 = (wid & 1) * 64;
    gemm_strip_f32p(h, D_, Wp_ee, m0, n0, acc, lane);
    store_strip_f32(hW, D_, m0, n0, acc, lane);
  } else {
    const int n0 = (wid - 1024) * 64;   // NN=16 rows
    gemm_strip_f32p(nuc_emb, D_, Wp_ne, 0, n0, acc, lane);
    store_strip_f32(hnW, D_, 0, n0, acc, lane);
  }
}

// ---------------------------------------------------------------------------
// K4: agg = m_ee + m_ne with on-the-fly Khatri-Rao B operand:
//   m_ee[b,i,d] = sum_{(j,k)} Aee[b,i,(j,k)] * ( hW[b,j,d] * Wrbf_ee[k,d] )
//   m_ne[b,i,d] = sum_{(a,k)} Ane[b,i,(a,k)] * ( hnW[a,d]  * Wrbf_ne[k,d] )
// j-outer loop: hW factor loaded once per j; the 16xD W_rbf block is preloaded
// into registers (each lane only needs its 8 half-specific k-rows x NT columns).
// Grid: B * (NE/16) * (D/64) = 1024 waves of 32; 1280 WMMA per wave.
// ---------------------------------------------------------------------------
__global__ void gnn_agg(const float* __restrict__ Aee, const float* __restrict__ Ane,
                        const float* __restrict__ hW, const float* __restrict__ hnW,
                        const float* __restrict__ Wrbf_ee_l, const float* __restrict__ Wrbf_ne_l,
                        float* __restrict__ agg) {
  const int wid = blockIdx.x;
  const int b  = wid >> 3;
  const int mt = (wid >> 1) & 3;
  const int nb = wid & 1;
  const int lane = threadIdx.x & 31;
  const int half = lane >> 4, l16 = lane & 15;
  const int i0 = mt * 16, n0 = nb * 64;

  const float* __restrict__ Ab  = Aee + (size_t)b * NE_ * KEE_;
  const float* __restrict__ An  = Ane + (size_t)b * NE_ * KNE_;
  const float* __restrict__ hWb = hW + (size_t)b * NE_ * D_;
  v8f acc[NT_] = {};

  float wk[NT_][8];   // per-lane slice of the 16xD W_rbf block (k rows it owns)

  // ---- electron -> electron, K = 1024 ----
  {
#pragma unroll
    for (int kk = 0; kk < 4; ++kk)
#pragma unroll
      for (int t = 0; t < NT_; ++t) {
        const int kb = kk * 4 + half * 2;
        wk[t][kk * 2 + 0] = Wrbf_ee_l[(size_t)(kb + 0) * D_ + n0 + t * 16 + l16];
        wk[t][kk * 2 + 1] = Wrbf_ee_l[(size_t)(kb + 1) * D_ + n0 + t * 16 + l16];
      }
    const float* __restrict__ arow = Ab + (size_t)(i0 + l16) * KEE_;
    for (int j = 0; j < NE_; ++j) {
      float hv[NT_];
#pragma unroll
      for (int t = 0; t < NT_; ++t) hv[t] = hWb[(size_t)j * D_ + n0 + t * 16 + l16];
#pragma unroll
      for (int kk = 0; kk < 4; ++kk) {
        const int ka = j * 16 + kk * 4 + half * 2;   // even
        const v2f a = *(const v2f*)&arow[ka];
#pragma unroll
        for (int t = 0; t < NT_; ++t) {
          v2f bb;
          bb.x = hv[t] * wk[t][kk * 2 + 0];
          bb.y = hv[t] * wk[t][kk * 2 + 1];
          acc[t] = __builtin_amdgcn_wmma_f32_16x16x4_f32(false, a, false, bb, (short)0,
                                                         acc[t], false, false);
        }
      }
    }
  }

  // ---- nucleus -> electron, K = 256 (B operand batch-independent) ----
  {
#pragma unroll
    for (int kk = 0; kk < 4; ++kk)
#pragma unroll
      for (int t = 0; t < NT_; ++t) {
        const int kb = kk * 4 + half * 2;
        wk[t][kk * 2 + 0] = Wrbf_ne_l[(size_t)(kb + 0) * D_ + n0 + t * 16 + l16];
        wk[t][kk * 2 + 1] = Wrbf_ne_l[(size_t)(kb + 1) * D_ + n0 + t * 16 + l16];
      }
    const float* __restrict__ arow = An + (size_t)(i0 + l16) * KNE_;
#pragma unroll
    for (int a_idx = 0; a_idx < NN_; ++a_idx) {
      float hv[NT_];
#pragma unroll
      for (int t = 0; t < NT_; ++t) hv[t] = hnW[(size_t)a_idx * D_ + n0 + t * 16 + l16];
#pragma unroll
      for (int kk = 0; kk < 4; ++kk) {
        const int ka = a_idx * 16 + kk * 4 + half * 2;
        const v2f a = *(const v2f*)&arow[ka];
#pragma unroll
        for (int t = 0; t < NT_; ++t) {
          v2f bb;
          bb.x = hv[t] * wk[t][kk * 2 + 0];
          bb.y = hv[t] * wk[t][kk * 2 + 1];
          acc[t] = __builtin_amdgcn_wmma_f32_16x16x4_f32(false, a, false, bb, (short)0,
                                                         acc[t], false, false);
        }
      }
    }
  }

  store_strip_f32(agg + (size_t)b * NE_ * D_, D_, i0, n0, acc, lane);
}

// ---------------------------------------------------------------------------
// K5a: tmp = tanh(agg @ W1[l])   (8192x128 @ 128x128)
// ---------------------------------------------------------------------------
__global__ void gnn_mlp1(const float* __restrict__ agg, const float* __restrict__ Wp1,
                         float* __restrict__ tmp) {
  const int wid = blockIdx.x;
  const int lane = threadIdx.x & 31;
  const int m0 = (wid >> 1) * 16;
  const int n0 = (wid & 1) * 64;
  v8f acc[NT_] = {};
  gemm_strip_f32p(agg, D_, Wp1, m0, n0, acc, lane);
#pragma unroll
  for (int t = 0; t < NT_; ++t)
#pragma unroll
    for (int ii = 0; ii < 8; ++ii) acc[t][ii] = tanhf(acc[t][ii]);
  store_strip_f32(tmp, D_, m0, n0, acc, lane);
}

// ---------------------------------------------------------------------------
// K5b: dst = h + tmp @ W2[l]  (residual preloaded into the accumulators)
// ---------------------------------------------------------------------------
__global__ void gnn_mlp2(const float* __restrict__ tmp, const float* __restrict__ Wp2,
                         const float* __restrict__ h, float* __restrict__ dst) {
  const int wid = blockIdx.x;
  const int lane = threadIdx.x & 31;
  const int m0 = (wid >> 1) * 16;
  const int n0 = (wid & 1) * 64;
  v8f acc[NT_];
  load_strip_f32(h, D_, m0, n0, acc, lane);         // residual -> C operand
  gemm_strip_f32p(tmp, D_, Wp2, m0, n0, acc, lane);
  store_strip_f32(dst, D_, m0, n0, acc, lane);
}

// ---------------------------------------------------------------------------
extern "C" void kernel_launch(void* const* d_in, const int* in_sizes, int n_in,
                              void* d_out, int out_size, void* d_ws, size_t ws_size,
                              hipStream_t stream) {
  (void)in_sizes; (void)n_in; (void)out_size; (void)ws_size;
  const float* r        = (const float*)d_in[0];
  const float* coords   = (const float*)d_in[1];
  const float* spin_emb = (const float*)d_in[2];
  const float* nuc_emb  = (const float*)d_in[3];
  const float* W_rbf_ee = (const float*)d_in[4];
  const float* W_rbf_ne = (const float*)d_in[5];
  const float* W_msg_ee = (const float*)d_in[6];
  const float* W_msg_ne = (const float*)d_in[7];
  const float* W1       = (const float*)d_in[8];
  const float* W2       = (const float*)d_in[9];
  float* out = (float*)d_out;

  // Workspace layout (floats), total ~14.9M floats (~60 MB)
  float* ws  = (float*)d_ws;
  float* Aee = ws;                                         // 8,388,608
  float* Ane = Aee + (size_t)B_ * NE_ * KEE_;              // 2,097,152
  float* h   = Ane + (size_t)B_ * NE_ * KNE_;              // 1,048,576
  float* hW  = h   + (size_t)B_ * NE_ * D_;                // 1,048,576
  float* hnW = hW  + (size_t)B_ * NE_ * D_;                // 2,048
  float* agg = hnW + (size_t)NN_ * D_;                     // 1,048,576
  float* tmp = agg + (size_t)B_ * NE_ * D_;                // 1,048,576
  float* Wp  = tmp + (size_t)B_ * NE_ * D_;                // 12 * 16384 = 196,608

  // Layer-independent precompute
  gnn_pack_w<<<(12 * WSZ_) / 256, 256, 0, stream>>>(W_msg_ee, W_msg_ne, W1, W2, Wp);
  gnn_feat_ee<<<B_ * NE_, 256, 0, stream>>>(r, Aee);
  gnn_feat_ne<<<B_ * NE_, 256, 0, stream>>>(r, coords, Ane);
  gnn_init_h<<<(B_ * NE_ * D_) / 256, 256, 0, stream>>>(spin_emb, h);

  for (int l = 0; l < L_; ++l) {
    const float* Wpl = Wp + (size_t)l * 4 * WSZ_;
    gnn_hw<<<1024 + 2, 32, 0, stream>>>(h, nuc_emb, Wpl + 0 * WSZ_, Wpl + 1 * WSZ_, hW, hnW);
    gnn_agg<<<B_ * 8, 32, 0, stream>>>(Aee, Ane, hW, hnW,
                                       W_rbf_ee + (size_t)l * NRBF_ * D_,
                                       W_rbf_ne + (size_t)l * NRBF_ * D_, agg);
    gnn_mlp1<<<1024, 32, 0, stream>>>(agg, Wpl + 2 * WSZ_, tmp);
    float* dst = (l == L_ - 1) ? out : h;
    gnn_mlp2<<<1024, 32, 0, stream>>>(tmp, Wpl + 3 * WSZ_, h, dst);
  }
}